// RNNDecoder_Angles_Single_SRU_87754771792273
// MI455X (gfx1250) — compile-verified
//
#include <hip/hip_runtime.h>
#include <hip/hip_bf16.h>

#define Bz   16
#define Lz   512
#define INz  512
#define HIDz 512
#define AAz  21

typedef float v2f __attribute__((ext_vector_type(2)));
typedef float v8f __attribute__((ext_vector_type(8)));

// ---------------------------------------------------------------------------
// GEMM tiling: block = 256 threads (8 waves), block tile 128x64.
// Wave w computes a 16x64 strip -> 4 WMMA accumulators, A fragment reused 4x.
// LDS tiles stored K-contiguous with padded stride 36 floats (144B, 16B
// aligned, conflict-free paired-bank float2 reads). Double-buffered with
// GLOBAL_LOAD_ASYNC_TO_LDS (ASYNCcnt) when available.
// ---------------------------------------------------------------------------
#define TMB 128
#define TNB 64
#define KC  32
#define LDP 36

#define HAS_ASYNC_LDS __has_builtin(__builtin_amdgcn_global_load_async_to_lds_b32)

typedef __attribute__((address_space(1))) int as1_int;
typedef __attribute__((address_space(3))) int as3_int;

__device__ __forceinline__ void cp_to_lds_b32(const float* g, float* l) {
#if HAS_ASYNC_LDS
    __builtin_amdgcn_global_load_async_to_lds_b32(
        (as1_int*)g, (as3_int*)l, 0, 0);
#else
    *l = *g;
#endif
}

__device__ __forceinline__ void wait_fill() {
#if HAS_ASYNC_LDS
#if __has_builtin(__builtin_amdgcn_s_wait_asynccnt)
    __builtin_amdgcn_s_wait_asynccnt(0);
#else
    asm volatile("s_wait_asynccnt 0x0" ::: "memory");
#endif
#endif
}

__device__ __forceinline__ void fill_tiles(
    const float* __restrict__ A, const float* __restrict__ B,
    float* As, float* Bs, int tid, int bm, int bn, int k0, int N, int K)
{
    // A tile: 128x32, global reads coalesced along K, LDS rows [m][kk]
    for (int i = tid; i < TMB * KC; i += 256) {
        int kk = i & (KC - 1);
        int m  = i >> 5;
        cp_to_lds_b32(&A[(size_t)(bm + m) * K + (k0 + kk)], &As[m * LDP + kk]);
    }
    // B tile: 32x64, global reads coalesced along N, stored transposed [n][kk]
    for (int i = tid; i < KC * TNB; i += 256) {
        int n  = i & (TNB - 1);
        int kk = i >> 6;
        cp_to_lds_b32(&B[(size_t)(k0 + kk) * N + (bn + n)], &Bs[n * LDP + kk]);
    }
#if !HAS_ASYNC_LDS
    // prefetch the K-block after next (emits global_prefetch_b8)
    if (k0 + KC < K) {
        __builtin_prefetch(&A[(size_t)(bm + (tid >> 5)) * K + (k0 + KC)], 0, 1);
        __builtin_prefetch(&B[(size_t)(k0 + KC + (tid >> 6)) * N + bn], 0, 1);
    }
#endif
}

__global__ __launch_bounds__(256) void gemm_f32_wmma_kernel(
    const float* __restrict__ A, const float* __restrict__ B,
    float* __restrict__ C, const float* __restrict__ bias,
    int M, int N, int K)
{
    __shared__ float As[2][TMB * LDP];
    __shared__ float Bs[2][TNB * LDP];

    const int bm   = blockIdx.y * TMB;
    const int bn   = blockIdx.x * TNB;
    const int tid  = threadIdx.x;
    const int wave = tid >> 5;
    const int lane = tid & 31;

    v8f acc[4] = {{}, {}, {}, {}};

    // prologue: tile 0 -> buffer 0
    fill_tiles(A, B, As[0], Bs[0], tid, bm, bn, 0, N, K);

    const int khalf = (lane >> 4) << 1;           // ISA A/B frag: lanes 16-31 = K+2
    const int mrow  = wave * 16 + (lane & 15);
    const int nrow  = lane & 15;

    int p = 0;
    for (int k0 = 0; k0 < K; k0 += KC) {
        wait_fill();
        __syncthreads();
        if (k0 + KC < K)
            fill_tiles(A, B, As[p ^ 1], Bs[p ^ 1], tid, bm, bn, k0 + KC, N, K);

        const float* Ap = &As[p][mrow * LDP];
        const float* Bp = &Bs[p][nrow * LDP];
#pragma unroll
        for (int kk = 0; kk < KC; kk += 4) {
            v2f a = *(const v2f*)&Ap[kk + khalf];
#pragma unroll
            for (int j = 0; j < 4; ++j) {
                v2f b = *(const v2f*)&Bp[j * 16 * LDP + kk + khalf];
                acc[j] = __builtin_amdgcn_wmma_f32_16x16x4_f32(
                             false, a, false, b, (short)0, acc[j], false, false);
            }
        }
        p ^= 1;
    }

    // C/D layout: VGPR i -> M = i (lanes 0-15) / M = 8+i (lanes 16-31), N = lane&15
    const int mbase = bm + wave * 16 + ((lane >> 4) << 3);
    const int ncb   = bn + (lane & 15);
#pragma unroll
    for (int j = 0; j < 4; ++j) {
        const int nc = ncb + 16 * j;
        const float bia = bias ? bias[nc] : 0.0f;
#pragma unroll
        for (int i = 0; i < 8; ++i)
            C[(size_t)(mbase + i) * N + nc] = acc[j][i] + bia;
    }
}

// ---------------------------------------------------------------------------
// [B,L,IN] -> X[l*B+b, IN]  (time-major rows so GEMM A rows are t*B+b)
// ---------------------------------------------------------------------------
__global__ __launch_bounds__(256) void transpose_in_kernel(
    const float* __restrict__ in, float* __restrict__ X)
{
    int row = blockIdx.x;            // l*B + b
    int l = row >> 4;
    int b = row & 15;
    for (int c = threadIdx.x; c < INz; c += 256)
        X[(size_t)row * INz + c] = in[((size_t)b * Lz + l) * INz + c];
}

// ---------------------------------------------------------------------------
// SRU elementwise scan: one thread per (b, dir, h) channel, serial over L.
// dir==1 runs time backwards (== reference's flip/scan/flip).
// ---------------------------------------------------------------------------
__global__ __launch_bounds__(256) void sru_scan_kernel(
    const float* __restrict__ U, const float* __restrict__ X,
    const float* __restrict__ v, const float* __restrict__ bprm,
    float* __restrict__ H, int k)
{
    int idx = blockIdx.x * 256 + threadIdx.x;    // 0..16383
    int h   = idx & (HIDz - 1);
    int dir = (idx >> 9) & 1;
    int b   = idx >> 10;

    const float vf = v[(0 * 2 + dir) * HIDz + h];
    const float vr = v[(1 * 2 + dir) * HIDz + h];
    const float bf = bprm[(0 * 2 + dir) * HIDz + h];
    const float br = bprm[(1 * 2 + dir) * HIDz + h];

    const size_t ncols   = (size_t)2 * HIDz * k;
    const size_t colbase = (size_t)(dir * HIDz + h) * k;
    const size_t hcol    = (size_t)dir * HIDz + h;

    float c = 0.0f;
    for (int s = 0; s < Lz; ++s) {
        int t = dir ? (Lz - 1 - s) : s;
        size_t row = (size_t)t * Bz + b;
        const float* u = U + row * ncols + colbase;
        float u0 = u[0], u1 = u[1], u2 = u[2];
        float res = (k == 4) ? u[3] : X[row * (2 * HIDz) + hcol];

        float f  = 1.0f / (1.0f + expf(-(u1 + vf * c + bf)));
        float c1 = f * c + (1.0f - f) * u0;
        float r  = 1.0f / (1.0f + expf(-(u2 + vr * c1 + br)));
        H[row * (2 * HIDz) + hcol] = r * c1 + (1.0f - r) * res;
        c = c1;
    }
}

// ---------------------------------------------------------------------------
// Heads: one wave per row (row = t*B + b). Fused small GEMMs + activations.
// ---------------------------------------------------------------------------
__global__ __launch_bounds__(32) void heads_kernel(
    const float* __restrict__ hlog, const float* __restrict__ hang,
    const float* __restrict__ w2l, const float* __restrict__ b2l,
    const float* __restrict__ w2m, const float* __restrict__ b2m,
    const float* __restrict__ w2k, const float* __restrict__ b2k,
    float* __restrict__ out_logits, float* __restrict__ out_means,
    float* __restrict__ out_kappas)
{
    const int row  = blockIdx.x;       // t*B + b
    const int b    = row & (Bz - 1);
    const int t    = row >> 4;
    const int lane = threadIdx.x;

    __shared__ float z[32];
    const float* hl = hlog + (size_t)row * HIDz;
    const float* ha = hang + (size_t)row * HIDz;

    float acc = 0.0f;
    if (lane < AAz) {
        for (int kk = 0; kk < HIDz; ++kk)
            acc += hl[kk] * w2l[kk * AAz + lane];
        acc += b2l[lane];
    }
    z[lane] = acc;
    __syncthreads();

    if (lane < AAz) {
        float m = -1e30f;
        for (int j = 0; j < AAz; ++j) m = fmaxf(m, z[j]);
        float s = 0.0f;
        for (int j = 0; j < AAz; ++j) s += expf(z[j] - m);
        out_logits[((size_t)b * Lz + t) * AAz + lane] = z[lane] - m - logf(s);
    }
    if (lane < 2) {
        float am = 0.0f, ak = 0.0f;
        for (int kk = 0; kk < HIDz; ++kk) {
            float hv = ha[kk];
            am += hv * w2m[kk * 2 + lane];
            ak += hv * w2k[kk * 2 + lane];
        }
        am += b2m[lane];
        ak += b2k[lane];
        out_means[((size_t)b * Lz + t) * 2 + lane] =
            tanhf(am) * 3.14159265358979323846f;
        float sp = (ak > 20.0f) ? ak : log1pf(expf(ak));
        out_kappas[((size_t)b * Lz + t) * 2 + lane] = 5.0f + sp;
    }
}

// ---------------------------------------------------------------------------
extern "C" void kernel_launch(void* const* d_in, const int* in_sizes, int n_in,
                              void* d_out, int out_size, void* d_ws, size_t ws_size,
                              hipStream_t stream) {
    const float* input = (const float*)d_in[0];
    const float* sru_w[6];
    const float* sru_v[6];
    const float* sru_b[6];
    for (int l = 0; l < 6; ++l) {
        sru_w[l] = (const float*)d_in[2 + 3 * l];
        sru_v[l] = (const float*)d_in[3 + 3 * l];
        sru_b[l] = (const float*)d_in[4 + 3 * l];
    }
    const float* fc1lw = (const float*)d_in[20];
    const float* fc1lb = (const float*)d_in[21];
    const float* fc2lw = (const float*)d_in[22];
    const float* fc2lb = (const float*)d_in[23];
    const float* fc1aw = (const float*)d_in[24];
    const float* fc1ab = (const float*)d_in[25];
    const float* fc2mw = (const float*)d_in[26];
    const float* fc2mb = (const float*)d_in[27];
    const float* fc2kw = (const float*)d_in[28];
    const float* fc2kb = (const float*)d_in[29];

    const int M = Lz * Bz;                    // 8192 rows
    float* ws = (float*)d_ws;
    float* Xbuf = ws;                         // 8192*1024
    float* Hbuf = Xbuf + (size_t)M * 1024;    // 8192*1024
    float* Ubuf = Hbuf + (size_t)M * 1024;    // 8192*4096 (reused for heads)
    float* hlog = Ubuf;                       // 8192*512
    float* hang = Ubuf + (size_t)M * 512;     // 8192*512

    // 1) [B,L,IN] -> time-major X
    transpose_in_kernel<<<M, 256, 0, stream>>>(input, Xbuf);

    // 2) SRU stack
    float* cur = Xbuf;
    for (int l = 0; l < 6; ++l) {
        const int din = (l == 0) ? INz : 2 * HIDz;
        const int k   = (l == 0) ? 4 : 3;
        const int N   = 2 * HIDz * k;         // 4096 or 3072
        dim3 grid(N / TNB, M / TMB);
        gemm_f32_wmma_kernel<<<grid, 256, 0, stream>>>(
            cur, sru_w[l], Ubuf, nullptr, M, N, din);
        float* nxt = (cur == Xbuf) ? Hbuf : Xbuf;
        sru_scan_kernel<<<(Bz * 2 * HIDz) / 256, 256, 0, stream>>>(
            Ubuf, cur, sru_v[l], sru_b[l], nxt, k);
        cur = nxt;
    }

    // 3) head FC1 GEMMs (bias fused)
    {
        dim3 grid(HIDz / TNB, M / TMB);       // 8 x 64
        gemm_f32_wmma_kernel<<<grid, 256, 0, stream>>>(
            cur, fc1lw, hlog, fc1lb, M, HIDz, 2 * HIDz);
        gemm_f32_wmma_kernel<<<grid, 256, 0, stream>>>(
            cur, fc1aw, hang, fc1ab, M, HIDz, 2 * HIDz);
    }

    // 4) fused small heads + activations
    float* out = (float*)d_out;
    float* out_logits = out;
    float* out_means  = out + (size_t)Bz * Lz * AAz;
    float* out_kappas = out_means + (size_t)Bz * Lz * 2;
    heads_kernel<<<M, 32, 0, stream>>>(hlog, hang, fc2lw, fc2lb, fc2mw, fc2mb,
                                       fc2kw, fc2kb, out_logits, out_means,
                                       out_kappas);
}